// HardNegativeMiningContrastiveLoss_16965120819304
// MI455X (gfx1250) — compile-verified
//
#include <hip/hip_runtime.h>
#include <math.h>

typedef float v2f __attribute__((ext_vector_type(2)));
typedef float v8f __attribute__((ext_vector_type(8)));

#define B_N 4096
#define D_K 512
#define TM 128
#define TN 128
#define KC 32
#define LDP 36   // padded LDS row stride (floats): float4-aligned, conflict-free

static __device__ __forceinline__ float block_reduce_sum(float v, float* sbuf) {
#pragma unroll
    for (int m = 16; m > 0; m >>= 1) v += __shfl_xor(v, m, 32);
    const int wave = threadIdx.x >> 5;
    const int lane = threadIdx.x & 31;
    if (lane == 0) sbuf[wave] = v;
    __syncthreads();
    float r = 0.0f;
    if (wave == 0) {
        r = (lane < 8) ? sbuf[lane] : 0.0f;
#pragma unroll
        for (int m = 16; m > 0; m >>= 1) r += __shfl_xor(r, m, 32);
        if (lane == 0) sbuf[0] = r;
    }
    __syncthreads();
    r = sbuf[0];
    __syncthreads();
    return r;
}

// ---------------------------------------------------------------- normalize
__global__ __launch_bounds__(256) void normalize_rows(
    const float* __restrict__ v, const float* __restrict__ t,
    float* __restrict__ vn, float* __restrict__ tn) {
    __shared__ float sbuf[8];
    const int b = blockIdx.x;
    const float* src;
    float* dst;
    if (b < B_N) { src = v + (size_t)b * D_K;         dst = vn + (size_t)b * D_K; }
    else         { src = t + (size_t)(b - B_N) * D_K; dst = tn + (size_t)(b - B_N) * D_K; }
    const int tid = threadIdx.x;
    const float x0 = src[tid];
    const float x1 = src[tid + 256];
    const float ss = block_reduce_sum(x0 * x0 + x1 * x1, sbuf);
    const float scale = 1.0f / fmaxf(sqrtf(ss), 1e-12f);
    dst[tid] = x0 * scale;
    dst[tid + 256] = x1 * scale;
}

// ---------------------------------------------------------------- GEMM (NT)
// C[BxB] = A[BxD] * Bm[BxD]^T, fp32 via V_WMMA_F32_16X16X4_F32
static __device__ __forceinline__ v8f wmma4(v2f a, v2f b, v8f c) {
    return __builtin_amdgcn_wmma_f32_16x16x4_f32(
        /*neg_a=*/false, a, /*neg_b=*/false, b,
        /*c_mod=*/(short)0, c, /*reuse_a=*/false, /*reuse_b=*/false);
}

__global__ __launch_bounds__(256) void gemm_nt(
    const float* __restrict__ A, const float* __restrict__ Bm, float* __restrict__ C) {
    __shared__ float As[TM * LDP];
    __shared__ float Bs[TN * LDP];

    const int tid  = threadIdx.x;
    const int bm   = blockIdx.x * TM;
    const int bn   = blockIdx.y * TN;
    const int wave = tid >> 5;
    const int lane = tid & 31;
    const int wm   = (wave & 3) * 32;   // 0,32,64,96
    const int wn   = (wave >> 2) * 64;  // 0,64
    const int lrow  = lane & 15;
    const int khalf = (lane >> 4) * 2;  // lanes 0-15 -> K{0,1}; 16-31 -> K{2,3}

    v8f c[2][4];
#pragma unroll
    for (int mi = 0; mi < 2; ++mi)
#pragma unroll
        for (int ni = 0; ni < 4; ++ni) c[mi][ni] = (v8f){};

    for (int k0 = 0; k0 < D_K; k0 += KC) {
        // cooperative panel load: 128 rows x 32 cols, 4x float4 per thread per panel
#pragma unroll
        for (int i = 0; i < 4; ++i) {
            const int idx = tid + i * 256;
            const int r   = idx >> 3;
            const int c4  = (idx & 7) * 4;
            const float4 av = *(const float4*)(A  + (size_t)(bm + r) * D_K + k0 + c4);
            *(float4*)(&As[r * LDP + c4]) = av;
            const float4 bv = *(const float4*)(Bm + (size_t)(bn + r) * D_K + k0 + c4);
            *(float4*)(&Bs[r * LDP + c4]) = bv;
        }
        __syncthreads();

#pragma unroll
        for (int k = 0; k < KC; k += 4) {
            const int kk = k + khalf;
            const v2f a0 = *(const v2f*)&As[(wm + lrow)      * LDP + kk];
            const v2f a1 = *(const v2f*)&As[(wm + 16 + lrow) * LDP + kk];
            const v2f b0 = *(const v2f*)&Bs[(wn + lrow)      * LDP + kk];
            const v2f b1 = *(const v2f*)&Bs[(wn + 16 + lrow) * LDP + kk];
            const v2f b2 = *(const v2f*)&Bs[(wn + 32 + lrow) * LDP + kk];
            const v2f b3 = *(const v2f*)&Bs[(wn + 48 + lrow) * LDP + kk];
            c[0][0] = wmma4(a0, b0, c[0][0]);
            c[0][1] = wmma4(a0, b1, c[0][1]);
            c[0][2] = wmma4(a0, b2, c[0][2]);
            c[0][3] = wmma4(a0, b3, c[0][3]);
            c[1][0] = wmma4(a1, b0, c[1][0]);
            c[1][1] = wmma4(a1, b1, c[1][1]);
            c[1][2] = wmma4(a1, b2, c[1][2]);
            c[1][3] = wmma4(a1, b3, c[1][3]);
        }
        __syncthreads();
    }

    // C/D layout: VGPR r -> M = r + 8*(lane>>4), N = lane&15
#pragma unroll
    for (int mi = 0; mi < 2; ++mi) {
#pragma unroll
        for (int ni = 0; ni < 4; ++ni) {
            const int rbase = bm + wm + mi * 16 + (lane >> 4) * 8;
            const int col   = bn + wn + ni * 16 + (lane & 15);
#pragma unroll
            for (int r = 0; r < 8; ++r)
                C[(size_t)(rbase + r) * B_N + col] = c[mi][ni][r];
        }
    }
}

// ---------------------------------------------------------------- row loss
#define INV_T 14.2857142857f   // 1/0.07

__global__ __launch_bounds__(256) void row_loss(
    const float* __restrict__ sim, const int* __restrict__ ids,
    float* __restrict__ rowLoss, float* __restrict__ rowPos, int writePos) {
    __shared__ float sbuf[8];
    const int a   = blockIdx.x;
    const int tid = threadIdx.x;
    const int ida = ids[a];
    const float* row = sim + (size_t)a * B_N;

    float rv[16];
    unsigned mbits = 0;
    float psum = 0.0f, pcnt = 0.0f;
#pragma unroll
    for (int i = 0; i < 16; ++i) {
        const int j = tid + i * 256;
        const float s = row[j];
        rv[i] = s;
        if (ids[j] == ida) { psum += s; pcnt += 1.0f; mbits |= (1u << i); }
    }
    const float psum_t = block_reduce_sum(psum, sbuf);
    const float pcnt_t = block_reduce_sum(pcnt, sbuf);
    const float mean_pos = psum_t / fmaxf(pcnt_t, 1.0f);
    const float lo = mean_pos - 0.2f;

    float neg = 0.0f;
#pragma unroll
    for (int i = 0; i < 16; ++i) {
        if (!((mbits >> i) & 1u)) {
            const float s = rv[i];
            const float w = (s < mean_pos && s > lo) ? 2.0f : 1.0f;
            neg += w * __expf(s * INV_T);
        }
    }
    const float neg_t = block_reduce_sum(neg, sbuf);

    float l = 0.0f;
    const bool valid = (pcnt_t > 0.5f) && (pcnt_t < (float)B_N - 0.5f);
    if (valid) {
#pragma unroll
        for (int i = 0; i < 16; ++i)
            if ((mbits >> i) & 1u)
                l += log1pf(neg_t * __expf(-rv[i] * INV_T));
    }
    const float l_t = block_reduce_sum(l, sbuf);
    if (tid == 0) {
        rowLoss[a] = l_t;
        if (writePos) rowPos[a] = pcnt_t;
    }
}

// ---------------------------------------------------------------- finalize
__global__ __launch_bounds__(256) void finalize_loss(
    const float* __restrict__ rl0, const float* __restrict__ rl1,
    const float* __restrict__ rp, float* __restrict__ out) {
    __shared__ float sbuf[8];
    float l = 0.0f, np = 0.0f;
    for (int i = threadIdx.x; i < B_N; i += 256) {
        l += rl0[i] + rl1[i];
        np += rp[i];
    }
    l  = block_reduce_sum(l, sbuf);
    np = block_reduce_sum(np, sbuf);
    if (threadIdx.x == 0)
        out[0] = (np > 0.0f) ? l / (2.0f * fmaxf(np, 1.0f)) : 0.0f;
}

// ---------------------------------------------------------------- launch
extern "C" void kernel_launch(void* const* d_in, const int* in_sizes, int n_in,
                              void* d_out, int out_size, void* d_ws, size_t ws_size,
                              hipStream_t stream) {
    const float* vfeat = (const float*)d_in[0];
    const float* tfeat = (const float*)d_in[1];
    const int*   ids   = (const int*)d_in[2];
    float*       out   = (float*)d_out;

    char* ws = (char*)d_ws;
    const size_t featB = (size_t)B_N * D_K * sizeof(float);   // 8 MB
    const size_t simB  = (size_t)B_N * B_N * sizeof(float);   // 64 MB
    float* vn  = (float*)(ws);
    float* tn  = (float*)(ws + featB);
    float* sim = (float*)(ws + 2 * featB);
    float* rl0 = (float*)(ws + 2 * featB + simB);
    float* rl1 = rl0 + B_N;
    float* rp  = rl1 + B_N;

    normalize_rows<<<2 * B_N, 256, 0, stream>>>(vfeat, tfeat, vn, tn);

    dim3 g(B_N / TM, B_N / TN);
    // v->t direction: sim = vn * tn^T
    gemm_nt<<<g, 256, 0, stream>>>(vn, tn, sim);
    row_loss<<<B_N, 256, 0, stream>>>(sim, ids, rl0, rp, 1);
    // t->v direction: reuse buffer, sim = tn * vn^T (== sim^T)
    gemm_nt<<<g, 256, 0, stream>>>(tn, vn, sim);
    row_loss<<<B_N, 256, 0, stream>>>(sim, ids, rl1, rp, 0);

    finalize_loss<<<1, 256, 0, stream>>>(rl0, rl1, rp, out);
}